// SpatialCrossAttention_PyTorch_34291018891926
// MI455X (gfx1250) — compile-verified
//
#include <hip/hip_runtime.h>

typedef __attribute__((ext_vector_type(2))) float v2f;
typedef __attribute__((ext_vector_type(8))) float v8f;

#define EMBED 256
#define NH 8
#define HD 32
#define NL 4
#define NPT 4
#define NQ 10000
#define NCAMS 6
#define NUM_VALUE 19560
#define MV (NCAMS * NUM_VALUE)   // 117360 rows for value projection
#define GK 256                   // K is 256 for every GEMM here
#define KCH 128                  // K-chunk staged in LDS
#define LDSK 132                 // padded LDS row stride (floats): 16B aligned, conflict-free

__device__ __constant__ int   c_lvlH[NL]     = {92, 46, 23, 12};
__device__ __constant__ int   c_lvlW[NL]     = {160, 80, 40, 20};
__device__ __constant__ int   c_lvlStart[NL] = {0, 14720, 18400, 19320};
__device__ __constant__ float c_invW[NL] = {1.0f/160.0f, 1.0f/80.0f, 1.0f/40.0f, 1.0f/20.0f};
__device__ __constant__ float c_invH[NL] = {1.0f/92.0f, 1.0f/46.0f, 1.0f/23.0f, 1.0f/12.0f};

// ---------------------------------------------------------------------------
// fp32 WMMA GEMM:  C[M x N] = A[M x 256] * B[256 x N] + bias (epilogue by MODE)
//  MODE 0: C = AB + bias
//  MODE 1: C = (AB + bias) * offset-normalizer (per output column)
//  MODE 3: C = AB + bias + R (residual)
// Block: 256 threads (8 wave32) computes a 64 x N stripe.
// N is a COMPILE-TIME parameter so the "second N-tile" predicate folds away:
// no EXEC manipulation may surround v_wmma (ISA: EXEC must be all 1s, WMMA is
// not skipped at EXEC==0). Each wave owns N-tile(s) {wave, wave+8} and 4
// M-tiles; B fragments are reused across the 4 M-tiles (4x less B traffic).
// ---------------------------------------------------------------------------
template <int MODE, int N>
__global__ __launch_bounds__(256)
void wmma_gemm_f32(const float* __restrict__ A, const float* __restrict__ Bw,
                   const float* __restrict__ bias, const float* __restrict__ R,
                   float* __restrict__ C, int M)
{
    constexpr bool HAS2 = (N >= 256);   // compile-time: second N-tile exists

    __shared__ float sA[64 * LDSK];     // 64 rows x 128-K chunk (padded)

    const int tid  = threadIdx.x;
    const int m0   = blockIdx.x * 64;
    const int wave = tid >> 5;
    const int lane = tid & 31;
    const int hl   = lane >> 4;         // half: K pair {0,1} vs {2,3}
    const int lm   = lane & 15;         // row (A) / column (B,C,D) within tile

    const int n0a = wave * 16;
    const int n0b = (wave + 8) * 16;

    v8f acc[4][2];
    #pragma unroll
    for (int mt = 0; mt < 4; ++mt)
        #pragma unroll
        for (int nt = 0; nt < 2; ++nt)
            acc[mt][nt] = (v8f){0.f,0.f,0.f,0.f,0.f,0.f,0.f,0.f};

    for (int kc = 0; kc < GK; kc += KCH) {
        __syncthreads();
        // cooperative float4 stage of the 64 x 128 A chunk
        for (int i = tid; i < 64 * (KCH / 4); i += 256) {
            const int r  = i >> 5;      // 32 float4 per row
            const int c4 = i & 31;
            const int gr = m0 + r;
            float4 vv = make_float4(0.f, 0.f, 0.f, 0.f);
            if (gr < M)
                vv = *(const float4*)(A + (long)gr * GK + kc + c4 * 4);
            *(float4*)&sA[r * LDSK + c4 * 4] = vv;
        }
        __syncthreads();

        for (int ks = 0; ks < KCH; ks += 4) {
            const int ka = ks + 2 * hl;   // within chunk
            const int kg = kc + ka;       // global K

            // gather all operand fragments first so loads can batch
            v2f a[4];
            #pragma unroll
            for (int mt = 0; mt < 4; ++mt) {
                a[mt].x = sA[(mt * 16 + lm) * LDSK + ka];
                a[mt].y = sA[(mt * 16 + lm) * LDSK + ka + 1];
            }
            v2f b0, b1;
            b0.x = Bw[(long)kg * N + n0a + lm];
            b0.y = Bw[(long)(kg + 1) * N + n0a + lm];
            if (HAS2) {
                b1.x = Bw[(long)kg * N + n0b + lm];
                b1.y = Bw[(long)(kg + 1) * N + n0b + lm];
            }

            #pragma unroll
            for (int mt = 0; mt < 4; ++mt) {
                acc[mt][0] = __builtin_amdgcn_wmma_f32_16x16x4_f32(
                                 false, a[mt], false, b0, (short)0, acc[mt][0], false, false);
                if (HAS2)
                    acc[mt][1] = __builtin_amdgcn_wmma_f32_16x16x4_f32(
                                     false, a[mt], false, b1, (short)0, acc[mt][1], false, false);
            }
        }
    }

    // epilogue
    #pragma unroll
    for (int nt = 0; nt < (HAS2 ? 2 : 1); ++nt) {
        const int n0 = (wave + 8 * nt) * 16;
        const int n  = n0 + lm;
        const float bv = bias[n];
        float scale = 1.0f;
        if (MODE == 1) {
            // column n encodes ((h*NL+l)*NP+p)*2+xy  ->  l = (n>>3)&3, xy = n&1
            const int l = (n >> 3) & 3;
            scale = (n & 1) ? c_invH[l] : c_invW[l];
        }
        #pragma unroll
        for (int mt = 0; mt < 4; ++mt) {
            #pragma unroll
            for (int r = 0; r < 8; ++r) {
                const int m = m0 + mt * 16 + r + 8 * hl;
                if (m < M) {
                    float v = (acc[mt][nt][r] + bv) * scale;
                    if (MODE == 3) v += R[(long)m * N + n];
                    C[(long)m * N + n] = v;
                }
            }
        }
    }
}

// ---------------------------------------------------------------------------
// softmax over 16 contiguous elements per (q, head)
// ---------------------------------------------------------------------------
__global__ __launch_bounds__(256)
void softmax16_kernel(const float* __restrict__ raw, float* __restrict__ aw, int n)
{
    const int i = blockIdx.x * blockDim.x + threadIdx.x;
    if (i >= n) return;
    const float* x = raw + (long)i * 16;
    float mx = -3.4e38f;
    #pragma unroll
    for (int j = 0; j < 16; ++j) mx = fmaxf(mx, x[j]);
    float e[16], s = 0.f;
    #pragma unroll
    for (int j = 0; j < 16; ++j) { e[j] = __expf(x[j] - mx); s += e[j]; }
    const float inv = 1.0f / s;
    float* y = aw + (long)i * 16;
    #pragma unroll
    for (int j = 0; j < 16; ++j) y[j] = e[j] * inv;
}

// ---------------------------------------------------------------------------
// MSDA + masked cam-mean. One block per query; wave = head, lane = channel.
// vproj is L2-resident (120 MB < 192 MB L2); corner gathers are branchless
// (clamp + validity weight, identical to reference) 128B coalesced loads.
// ---------------------------------------------------------------------------
__global__ __launch_bounds__(256)
void msda_kernel(const float* __restrict__ vproj, const float* __restrict__ off,
                 const float* __restrict__ aw, const float* __restrict__ ref,
                 const float* __restrict__ mask, float* __restrict__ slots)
{
    __shared__ float s_off[EMBED];          // (h,l,p,2)
    __shared__ float s_aw[NH * NL * NPT];   // (h,l,p)
    __shared__ float s_ref[NCAMS * 8];      // (c,p,2)
    __shared__ float s_mask[8];

    const int q   = blockIdx.x;
    const int tid = threadIdx.x;

    s_off[tid] = off[(long)q * EMBED + tid];
    if (tid < 128) s_aw[tid] = aw[(long)q * 128 + tid];
    if (tid < NCAMS * 8) {
        const int c = tid >> 3, r = tid & 7;
        s_ref[tid] = ref[((long)c * NQ + q) * 8 + r];
    }
    if (tid < NCAMS) s_mask[tid] = mask[(long)tid * NQ + q];
    __syncthreads();

    const int h    = tid >> 5;   // head
    const int lane = tid & 31;   // channel within head (hd == wave32)

    float count = 0.f;
    #pragma unroll
    for (int c = 0; c < NCAMS; ++c) count += s_mask[c];
    count = fmaxf(count, 1.0f);

    float acc = 0.f;
    for (int c = 0; c < NCAMS; ++c) {
        if (s_mask[c] == 0.0f) continue;   // wave-uniform skip
        const float* vb = vproj + (long)c * NUM_VALUE * EMBED + h * HD + lane;
        #pragma unroll
        for (int l = 0; l < NL; ++l) {
            const int H = c_lvlH[l], W = c_lvlW[l];
            const float* vl = vb + (long)c_lvlStart[l] * EMBED;
            #pragma unroll
            for (int p = 0; p < NPT; ++p) {
                const int oi = ((h * NL + l) * NPT + p) * 2;
                const float lx = s_ref[c * 8 + p * 2 + 0] + s_off[oi + 0];
                const float ly = s_ref[c * 8 + p * 2 + 1] + s_off[oi + 1];
                const float ix = lx * (float)W - 0.5f;
                const float iy = ly * (float)H - 0.5f;
                const float fx = floorf(ix), fy = floorf(iy);
                const float wx1 = ix - fx, wx0 = 1.0f - wx1;
                const float wy1 = iy - fy, wy0 = 1.0f - wy1;
                const int x0 = (int)fx, y0 = (int)fy;
                const int x1 = x0 + 1,  y1 = y0 + 1;
                // validity flags (zero-padding outside), clamped gather indices
                const float vldx0 = (x0 >= 0 && x0 < W) ? 1.f : 0.f;
                const float vldx1 = (x1 >= 0 && x1 < W) ? 1.f : 0.f;
                const float vldy0 = (y0 >= 0 && y0 < H) ? 1.f : 0.f;
                const float vldy1 = (y1 >= 0 && y1 < H) ? 1.f : 0.f;
                const int x0c = min(max(x0, 0), W - 1);
                const int x1c = min(max(x1, 0), W - 1);
                const int y0c = min(max(y0, 0), H - 1);
                const int y1c = min(max(y1, 0), H - 1);
                float sv;
                sv  = (wx0 * wy0 * vldx0 * vldy0) * vl[(long)(y0c * W + x0c) * EMBED];
                sv += (wx1 * wy0 * vldx1 * vldy0) * vl[(long)(y0c * W + x1c) * EMBED];
                sv += (wx0 * wy1 * vldx0 * vldy1) * vl[(long)(y1c * W + x0c) * EMBED];
                sv += (wx1 * wy1 * vldx1 * vldy1) * vl[(long)(y1c * W + x1c) * EMBED];
                acc += sv * s_aw[(h * NL + l) * NPT + p];
            }
        }
    }
    slots[(long)q * EMBED + h * HD + lane] = acc / count;
}

// ---------------------------------------------------------------------------
// launcher
// ---------------------------------------------------------------------------
extern "C" void kernel_launch(void* const* d_in, const int* in_sizes, int n_in,
                              void* d_out, int out_size, void* d_ws, size_t ws_size,
                              hipStream_t stream)
{
    (void)in_sizes; (void)n_in; (void)out_size; (void)ws_size;

    const float* query   = (const float*)d_in[0];
    // d_in[1] = key : unused by the reference
    const float* value   = (const float*)d_in[2];   // (6,19560,1,256) == (117360,256)
    const float* ref     = (const float*)d_in[3];   // (6,1,10000,4,2)
    const float* bevmask = (const float*)d_in[4];   // (6,1,10000)
    // d_in[5] = spatial_shapes (compile-time constants here)
    const float* W_value = (const float*)d_in[6];
    const float* b_value = (const float*)d_in[7];
    const float* W_off   = (const float*)d_in[8];
    const float* b_off   = (const float*)d_in[9];
    const float* W_attn  = (const float*)d_in[10];
    const float* b_attn  = (const float*)d_in[11];
    const float* W_out   = (const float*)d_in[12];
    const float* b_out   = (const float*)d_in[13];

    float* out = (float*)d_out;

    float* ws    = (float*)d_ws;
    float* vproj = ws;                                  // 117360*256
    float* off   = vproj + (long)MV * EMBED;            // 10000*256
    float* attn  = off   + (long)NQ * EMBED;            // 10000*128 (pre-softmax)
    float* aw    = attn  + (long)NQ * 128;              // 10000*128
    float* slots = aw    + (long)NQ * 128;              // 10000*256

    const int gbM = (MV + 63) / 64;     // 1834
    const int gbQ = (NQ + 63) / 64;     // 157

    // 1) value projection: vproj = value @ W_value + b_value
    wmma_gemm_f32<0, 256><<<gbM, 256, 0, stream>>>(value, W_value, b_value, nullptr,
                                                   vproj, MV);
    // 2) sampling offsets (query is cam-broadcast -> compute once), normalizer fused
    wmma_gemm_f32<1, 256><<<gbQ, 256, 0, stream>>>(query, W_off, b_off, nullptr,
                                                   off, NQ);
    // 3) attention logits
    wmma_gemm_f32<0, 128><<<gbQ, 256, 0, stream>>>(query, W_attn, b_attn, nullptr,
                                                   attn, NQ);
    // 4) softmax over the 16 (level,point) logits per (q, head)
    softmax16_kernel<<<(NQ * NH + 255) / 256, 256, 0, stream>>>(attn, aw, NQ * NH);
    // 5) deformable sampling + masked cam-mean (deterministic, no atomics)
    msda_kernel<<<NQ, 256, 0, stream>>>(vproj, off, aw, ref, bevmask, slots);
    // 6) output projection + bias + residual
    wmma_gemm_f32<3, 256><<<gbQ, 256, 0, stream>>>(slots, W_out, b_out, query,
                                                   out, NQ);
}